// Convolve_42185168781626
// MI455X (gfx1250) — compile-verified
//
#include <hip/hip_runtime.h>
#include <hip/hip_bf16.h>

#define N_SRC   100000
#define N_DST   100000
#define N_EDGES 1250000
#define D_FEAT  64
#define HIDDEN  64
#define OUTF    128
#define KCAT    128   // HIDDEN + D_FEAT

typedef __attribute__((ext_vector_type(2))) float v2f;
typedef __attribute__((ext_vector_type(8))) float v8f;

// ---------------------------------------------------------------------------
// Zero scratch accumulators (vs, deg, sumsq) -- re-run every call so graph
// replays are deterministic.
// ---------------------------------------------------------------------------
__global__ void zero_kernel(float* __restrict__ p, size_t n) {
    size_t i = (size_t)blockIdx.x * blockDim.x + threadIdx.x;
    size_t stride = (size_t)gridDim.x * blockDim.x;
    for (; i < n; i += stride) p[i] = 0.0f;
}

// ---------------------------------------------------------------------------
// fc1: hs = relu(h_src @ W1 + b1).   One wave -> one 16x16 tile, K=64.
// Block = 128 threads (4 waves) covering the 4 N-tiles of one M-tile.
// W1 (64x64 f32 = 16KB) staged in LDS.
// ---------------------------------------------------------------------------
__global__ __launch_bounds__(128) void fc1_kernel(
    const float* __restrict__ h_src, const float* __restrict__ W1,
    const float* __restrict__ b1, float* __restrict__ hs) {
    __shared__ float lW[D_FEAT * HIDDEN];

    const int tid = threadIdx.x;
    for (int i = tid; i < D_FEAT * HIDDEN; i += 128) lW[i] = W1[i];
    __syncthreads();

    const int wave = tid >> 5;
    const int lane = tid & 31;
    const int half = lane >> 4;     // 0: K=0,1   1: K=2,3
    const int l15  = lane & 15;
    const int m0   = blockIdx.x * 16;
    const int n0   = wave * 16;
    const int col  = n0 + l15;

    const float* arow = h_src + (size_t)(m0 + l15) * D_FEAT + half * 2;

    v8f c = {0.f, 0.f, 0.f, 0.f, 0.f, 0.f, 0.f, 0.f};
#pragma unroll
    for (int k = 0; k < D_FEAT; k += 4) {
        v2f a = *(const v2f*)(arow + k);
        v2f b;
        b.x = lW[(k + half * 2    ) * HIDDEN + col];
        b.y = lW[(k + half * 2 + 1) * HIDDEN + col];
        c = __builtin_amdgcn_wmma_f32_16x16x4_f32(
                /*neg_a=*/false, a, /*neg_b=*/false, b,
                /*c_mod=*/(short)0, c, /*reuse_a=*/false, /*reuse_b=*/false);
    }

    const float bias = b1[col];
#pragma unroll
    for (int i = 0; i < 8; ++i) {
        int r = m0 + i + half * 8;
        float v = c[i] + bias;
        hs[(size_t)r * HIDDEN + col] = v > 0.f ? v : 0.f;
    }
}

// ---------------------------------------------------------------------------
// Edge phase: vs[dst,:] += hs[src,:] * w ;  deg[dst] += w.
// One wave per edge, 2 feats per lane (float2 gather, 2x f32 atomic scatter).
// Prefetch the next edge's source row (global_prefetch_b8 path).
// ---------------------------------------------------------------------------
__global__ __launch_bounds__(256) void edge_kernel(
    const float* __restrict__ hs, const float* __restrict__ ew,
    const int* __restrict__ src, const int* __restrict__ dst,
    float* __restrict__ vs, float* __restrict__ deg) {
    const int lane = threadIdx.x & 31;
    const long long e = (long long)blockIdx.x * 8 + (threadIdx.x >> 5);
    if (e >= N_EDGES) return;

    if (e + 8 < N_EDGES) {  // prefetch source row for the edge 8 ahead
        int s2 = src[e + 8];
        __builtin_prefetch(hs + (size_t)s2 * HIDDEN + lane * 2, 0, 1);
    }

    const float w = ew[e];
    const int s = src[e];
    const int d = dst[e];

    v2f hv = *(const v2f*)(hs + (size_t)s * HIDDEN + lane * 2);
    float* vrow = vs + (size_t)d * HIDDEN + lane * 2;
    atomicAdd(vrow    , hv.x * w);
    atomicAdd(vrow + 1, hv.y * w);
    if (lane == 0) atomicAdd(deg + d, w);
}

// ---------------------------------------------------------------------------
// fc2: out = relu(concat(vs/max(deg,1), h_dst) @ W2 + b2); accumulate sum of
// squares for the global L2 norm. Block = 256 threads (8 waves = 8 N-tiles of
// one 16-row M-tile). X-tile (16x128, 8KB) + W2 half (64x128, 32KB) in LDS.
// ---------------------------------------------------------------------------
__global__ __launch_bounds__(256) void fc2_kernel(
    const float* __restrict__ vs, const float* __restrict__ deg,
    const float* __restrict__ h_dst, const float* __restrict__ W2,
    const float* __restrict__ b2, float* __restrict__ out,
    float* __restrict__ sumsq) {
    __shared__ float lW[64 * OUTF];    // 32 KB: one K-half of W2
    __shared__ float lX[16 * KCAT];    // 8 KB: concat input tile

    const int tid = threadIdx.x;
    const int m0  = blockIdx.x * 16;

    // Stage X tile: x[r][k] = k<64 ? vs[r][k]/max(deg[r],1) : h_dst[r][k-64]
    for (int i = tid; i < 16 * KCAT; i += 256) {
        int r = i >> 7;
        int k = i & 127;
        int grow = m0 + r;
        float x;
        if (k < HIDDEN) {
            float dg = deg[grow];
            dg = dg > 1.f ? dg : 1.f;
            x = vs[(size_t)grow * HIDDEN + k] / dg;
        } else {
            x = h_dst[(size_t)grow * D_FEAT + (k - HIDDEN)];
        }
        lX[i] = x;
    }

    const int wave = tid >> 5;
    const int lane = tid & 31;
    const int half = lane >> 4;
    const int l15  = lane & 15;
    const int n0   = wave * 16;
    const int col  = n0 + l15;

    v8f c = {0.f, 0.f, 0.f, 0.f, 0.f, 0.f, 0.f, 0.f};
    for (int phase = 0; phase < 2; ++phase) {
        __syncthreads();  // lX staged / previous lW consumed
        for (int i = tid; i < 64 * OUTF; i += 256)
            lW[i] = W2[(size_t)phase * 64 * OUTF + i];
        __syncthreads();

        const float* xrow = lX + (size_t)l15 * KCAT + phase * 64 + half * 2;
#pragma unroll
        for (int k = 0; k < 64; k += 4) {
            v2f a = *(const v2f*)(xrow + k);
            v2f b;
            b.x = lW[(k + half * 2    ) * OUTF + col];
            b.y = lW[(k + half * 2 + 1) * OUTF + col];
            c = __builtin_amdgcn_wmma_f32_16x16x4_f32(
                    false, a, false, b, (short)0, c, false, false);
        }
    }

    const float bias = b2[col];
    float local = 0.f;
#pragma unroll
    for (int i = 0; i < 8; ++i) {
        int r = m0 + i + half * 8;
        float v = c[i] + bias;
        v = v > 0.f ? v : 0.f;
        out[(size_t)r * OUTF + col] = v;
        local += v * v;
    }
    // wave32 butterfly reduction, then one atomic per wave
    for (int off = 16; off > 0; off >>= 1)
        local += __shfl_xor(local, off, 32);
    if (lane == 0) atomicAdd(sumsq, local);
}

// ---------------------------------------------------------------------------
// Global L2 normalization: out *= 1/sqrt(sumsq). float4 per thread.
// ---------------------------------------------------------------------------
__global__ __launch_bounds__(256) void scale_kernel(
    float* __restrict__ out, const float* __restrict__ sumsq) {
    const float s = 1.0f / sqrtf(*sumsq);
    size_t i4 = ((size_t)blockIdx.x * blockDim.x + threadIdx.x) * 4;
    if (i4 < (size_t)N_DST * OUTF) {
        float4 v = *(float4*)(out + i4);
        v.x *= s; v.y *= s; v.z *= s; v.w *= s;
        *(float4*)(out + i4) = v;
    }
}

// ---------------------------------------------------------------------------
extern "C" void kernel_launch(void* const* d_in, const int* in_sizes, int n_in,
                              void* d_out, int out_size, void* d_ws, size_t ws_size,
                              hipStream_t stream) {
    const float* h_src = (const float*)d_in[0];
    const float* h_dst = (const float*)d_in[1];
    const float* ew    = (const float*)d_in[2];
    const float* W1    = (const float*)d_in[3];
    const float* b1    = (const float*)d_in[4];
    const float* W2    = (const float*)d_in[5];
    const float* b2    = (const float*)d_in[6];
    const int*   src   = (const int*)d_in[7];
    const int*   dst   = (const int*)d_in[8];
    float* out = (float*)d_out;

    // Workspace layout: hs | vs | deg | sumsq
    float* hs   = (float*)d_ws;
    float* vs   = hs + (size_t)N_SRC * HIDDEN;
    float* degp = vs + (size_t)N_DST * HIDDEN;
    float* ssq  = degp + N_DST;

    // 1) zero accumulators (vs, deg, sumsq are contiguous)
    size_t nz = (size_t)N_DST * HIDDEN + N_DST + 1;
    zero_kernel<<<2048, 256, 0, stream>>>(vs, nz);

    // 2) fc1: 6250 M-tiles x (4 waves covering N)
    fc1_kernel<<<N_SRC / 16, 128, 0, stream>>>(h_src, W1, b1, hs);

    // 3) edge scatter: 8 edges per 256-thread block
    edge_kernel<<<(N_EDGES + 7) / 8, 256, 0, stream>>>(hs, ew, src, dst, vs, degp);

    // 4) fc2 + sum of squares
    fc2_kernel<<<N_DST / 16, 256, 0, stream>>>(vs, degp, h_dst, W2, b2, out, ssq);

    // 5) global L2 normalize
    size_t total4 = ((size_t)N_DST * OUTF) / 4;
    scale_kernel<<<(unsigned)((total4 + 255) / 256), 256, 0, stream>>>(out, ssq);
}